// PAdicStructureLoss_70497593196922
// MI455X (gfx1250) — compile-verified
//
#include <hip/hip_runtime.h>
#include <hip/hip_bf16.h>

typedef __attribute__((ext_vector_type(2))) float v2f;
typedef __attribute__((ext_vector_type(8))) float v8f;

#define N_TOK   8192
#define DIM     256
#define TAB_N   81
#define TAB_SZ  (TAB_N * TAB_N)   // 6561 floats = 26244 B in LDS
#define MAX_IDX 63
#define EPSF    1e-10
#define ST      256                // super-tiles (32x32) per dimension
#define NPAIRS  (ST * (ST + 1) / 2)  // 32896 upper-triangular super-tile pairs

// d_ws layout:
//   [0,      32768) : float sq[8192]      row squared norms
//   [32768,  65536) : int   cidx[8192]    clamped indices
//   [65536,  65536+28): float acc[7]      0:Se 1:Se2 2:St 3:St2 4:Set 5:maxE(bits) 6:maxT(bits)

__device__ __forceinline__ float waveSum(float v) {
#pragma unroll
  for (int o = 16; o > 0; o >>= 1) v += __shfl_xor(v, o, 32);
  return v;
}
__device__ __forceinline__ float waveMax(float v) {
#pragma unroll
  for (int o = 16; o > 0; o >>= 1) v = fmaxf(v, __shfl_xor(v, o, 32));
  return v;
}

__global__ void padic_init(float* acc) {
  if (threadIdx.x < 7) acc[threadIdx.x] = 0.0f;  // 0.0f bits == 0u (uint-max identity)
}

// grid: 1024 blocks x 256 threads. Each wave computes one row norm; first 8192
// threads overall also clamp one index.
__global__ void padic_prep(const float* __restrict__ X,
                           const int* __restrict__ idx,
                           float* __restrict__ sq,
                           int* __restrict__ cidx) {
  const int lane = threadIdx.x & 31;
  const int wave = threadIdx.x >> 5;
  const int row  = blockIdx.x * 8 + wave;
  float s = 0.0f;
  const float* xr = X + (size_t)row * DIM;
#pragma unroll
  for (int k = 0; k < DIM / 32; ++k) {
    float v = xr[lane + 32 * k];
    s = fmaf(v, v, s);
  }
  s = waveSum(s);
  if (lane == 0) sq[row] = s;

  const int t = blockIdx.x * blockDim.x + threadIdx.x;
  if (t < N_TOK) cidx[t] = min(idx[t], MAX_IDX);
}

// cumulative pair count at super-tile row i (row-major upper triangle)
__device__ __forceinline__ int triStart(int i) { return i * ST - (i * (i - 1)) / 2; }

// grid: NPAIRS/8 blocks x 256 threads (8 waves). Wave w handles upper-tri
// super-tile pair p = blockIdx.x*8 + w -> a 32x32 Gram region computed as
// 2x2 register-blocked 16x16 WMMA tiles (4 independent v_wmma per K-step,
// each A/B fragment reused twice -> 8 FLOP/B from L2).
__global__ void __launch_bounds__(256)
padic_tiles(const float* __restrict__ X,
            const float* __restrict__ sq,
            const int* __restrict__ cidx,
            const float* __restrict__ tab,
            float* __restrict__ acc) {
  __shared__ float ldsTab[TAB_SZ];
  __shared__ float red[7][8];

  for (int t = threadIdx.x; t < TAB_SZ; t += 256) ldsTab[t] = tab[t];
  __syncthreads();

  const int lane = threadIdx.x & 31;
  const int wave = threadIdx.x >> 5;
  const int p    = blockIdx.x * 8 + wave;   // NPAIRS % 8 == 0 -> always valid

  // decode linear pair index -> (iT, jT), iT <= jT
  int iT = (int)((513.0 - sqrt(513.0 * 513.0 - 8.0 * (double)p)) * 0.5);
  while (triStart(iT + 1) <= p) ++iT;
  while (triStart(iT) > p) --iT;
  const int jT = iT + (p - triStart(iT));
  const int i0 = iT * 32;
  const int j0 = jT * 32;

  // fp32 WMMA fragment addressing (16x4 A / 4x16 B):
  //   lane holds row (lane&15), K = 2*(lane>>4) + {0,1}  -> one aligned v2f load
  const int   koff = 2 * (lane >> 4);
  const float* aP0 = X + (size_t)(i0 + (lane & 15)) * DIM + koff;
  const float* aP1 = aP0 + (size_t)16 * DIM;
  const float* bP0 = X + (size_t)(j0 + (lane & 15)) * DIM + koff;
  const float* bP1 = bP0 + (size_t)16 * DIM;

  v8f c00 = {}, c01 = {}, c10 = {}, c11 = {};
#pragma unroll 4
  for (int k = 0; k < DIM; k += 4) {
    v2f a0 = *(const v2f*)(aP0 + k);
    v2f a1 = *(const v2f*)(aP1 + k);
    v2f b0 = *(const v2f*)(bP0 + k);
    v2f b1 = *(const v2f*)(bP1 + k);
    c00 = __builtin_amdgcn_wmma_f32_16x16x4_f32(false, a0, false, b0, (short)0, c00, false, false);
    c01 = __builtin_amdgcn_wmma_f32_16x16x4_f32(false, a0, false, b1, (short)0, c01, false, false);
    c10 = __builtin_amdgcn_wmma_f32_16x16x4_f32(false, a1, false, b0, (short)0, c10, false, false);
    c11 = __builtin_amdgcn_wmma_f32_16x16x4_f32(false, a1, false, b1, (short)0, c11, false, false);
  }

  // C layout: VGPR v, lanes 0-15 -> M=v, N=lane; lanes 16-31 -> M=v+8, N=lane-16
  const int jc0   = j0 + (lane & 15);
  const int jc1   = jc0 + 16;
  const float sqj0 = sq[jc0];
  const float sqj1 = sq[jc1];
  const int  idxj0 = cidx[jc0];
  const int  idxj1 = cidx[jc1];
  const int  mBase = i0 + 8 * (lane >> 4);

  float se = 0.f, se2 = 0.f, st = 0.f, st2 = 0.f, setv = 0.f;
  float me = 0.f, mt = 0.f;

#pragma unroll
  for (int ti = 0; ti < 2; ++ti) {
#pragma unroll
    for (int tj = 0; tj < 2; ++tj) {
      const v8f& cc  = (ti == 0) ? ((tj == 0) ? c00 : c01) : ((tj == 0) ? c10 : c11);
      const int   j   = (tj == 0) ? jc0 : jc1;
      const float sqj = (tj == 0) ? sqj0 : sqj1;
      const int idxj  = (tj == 0) ? idxj0 : idxj1;
#pragma unroll
      for (int v = 0; v < 8; ++v) {
        const int i = mBase + 16 * ti + v;
        const float g = cc[v];
        float d2 = fmaxf(sq[i] + sqj - 2.0f * g, 0.0f);
        const float dist = (d2 > 0.0f) ? sqrtf(d2) : 0.0f;
        const float tgt  = ldsTab[cidx[i] * TAB_N + idxj];
        // symmetry weight: upper triangle counted twice, diagonal once,
        // lower triangle (only occurs inside diagonal super-tiles) zero.
        const float w = (i < j) ? 2.0f : ((i == j) ? 1.0f : 0.0f);
        const float wd = w * dist;
        const float wt = w * tgt;
        se  += wd;
        se2  = fmaf(wd, dist, se2);
        st  += wt;
        st2  = fmaf(wt, tgt, st2);
        setv = fmaf(wd, tgt, setv);
        me   = fmaxf(me, dist);   // duplicates/zero-weight values can't change a max
        mt   = fmaxf(mt, tgt);
      }
    }
  }

  // wave reduce -> LDS -> block reduce -> global atomics
  se = waveSum(se); se2 = waveSum(se2); st = waveSum(st);
  st2 = waveSum(st2); setv = waveSum(setv);
  me = waveMax(me); mt = waveMax(mt);
  if (lane == 0) {
    red[0][wave] = se;  red[1][wave] = se2; red[2][wave] = st;
    red[3][wave] = st2; red[4][wave] = setv;
    red[5][wave] = me;  red[6][wave] = mt;
  }
  __syncthreads();
  if (threadIdx.x < 7) {
    const int q = threadIdx.x;
    if (q < 5) {
      float s = 0.f;
#pragma unroll
      for (int w = 0; w < 8; ++w) s += red[q][w];
      atomicAdd(&acc[q], s);
    } else {
      float m = 0.f;
#pragma unroll
      for (int w = 0; w < 8; ++w) m = fmaxf(m, red[q][w]);
      atomicMax((unsigned int*)&acc[q], __float_as_uint(m));  // values >= 0
    }
  }
}

__global__ void padic_finalize(const float* __restrict__ acc,
                               float* __restrict__ out) {
  const double M   = (double)N_TOK * (double)N_TOK;
  const double Se  = acc[0], Se2 = acc[1];
  const double St  = acc[2], St2 = acc[3], Set = acc[4];
  const double Me  = (double)__uint_as_float(((const unsigned int*)acc)[5]) + EPSF;
  const double Mt  = (double)__uint_as_float(((const unsigned int*)acc)[6]) + EPSF;

  const double cov  = Set - Se * St / M;
  double vare = Se2 - Se * Se / M; if (vare < 0.0) vare = 0.0;
  double vart = St2 - St * St / M; if (vart < 0.0) vart = 0.0;

  const double num = cov / (Me * Mt);
  const double den = sqrt(vare * vart) / (Me * Mt) + EPSF;
  out[0] = (float)(1.0 - num / den);
}

extern "C" void kernel_launch(void* const* d_in, const int* in_sizes, int n_in,
                              void* d_out, int out_size, void* d_ws, size_t ws_size,
                              hipStream_t stream) {
  const float* X   = (const float*)d_in[0];   // (8,1024,256) fp32
  const int*   idx = (const int*)d_in[1];     // (8,1024) int32
  const float* tab = (const float*)d_in[2];   // (81,81) fp32
  float* out = (float*)d_out;

  char*  ws   = (char*)d_ws;
  float* sq   = (float*)ws;
  int*   cidx = (int*)(ws + 32768);
  float* acc  = (float*)(ws + 65536);

  padic_init<<<1, 32, 0, stream>>>(acc);
  padic_prep<<<N_TOK / 8, 256, 0, stream>>>(X, idx, sq, cidx);
  padic_tiles<<<NPAIRS / 8, 256, 0, stream>>>(X, sq, cidx, tab, acc);
  padic_finalize<<<1, 1, 0, stream>>>(acc, out);
}